// GIN_1288490189415
// MI455X (gfx1250) — compile-verified
//
#include <hip/hip_runtime.h>

#define DHID 256

typedef __attribute__((ext_vector_type(2))) float v2f;
typedef __attribute__((ext_vector_type(8))) float v8f;

// ---------------------------------------------------------------------------
// agg = src  (self term of GIN: h2 = h + sum_neighbors)
// ---------------------------------------------------------------------------
__global__ void copy_f4_kernel(const float4* __restrict__ s, float4* __restrict__ d, long n4) {
    long i = (long)blockIdx.x * blockDim.x + threadIdx.x;
    if (i < n4) d[i] = s[i];
}

// ---------------------------------------------------------------------------
// agg[dst[e]] += h[src[e]]   — one (edge, 4-float chunk) per thread.
// h and agg are both ~L2-resident (102MB each vs 192MB L2), so the random
// gather + atomic scatter mostly runs at L2 speed, not HBM.
// ---------------------------------------------------------------------------
template <int D>
__global__ void scatter_add_kernel(const float* __restrict__ h,
                                   const int* __restrict__ src,
                                   const int* __restrict__ dst,
                                   float* __restrict__ agg, int nE) {
    constexpr int TPE = D / 4;                 // threads per edge (float4 chunks)
    long gid = (long)blockIdx.x * blockDim.x + threadIdx.x;
    int e = (int)(gid / TPE);
    int c = (int)(gid % TPE);
    if (e >= nE) return;
    int s = src[e];
    int d = dst[e];
    float4 v = *(const float4*)(h + (size_t)s * D + c * 4);
    float* base = agg + (size_t)d * D + c * 4;
    atomicAdd(base + 0, v.x);
    atomicAdd(base + 1, v.y);
    atomicAdd(base + 2, v.z);
    atomicAdd(base + 3, v.w);
}

// ---------------------------------------------------------------------------
// out = [relu](A @ W + bias),  A: nRows x K (row-major), W: K x 256, out: nRows x 256
// Block = 256 threads = 8 waves computing a 64-row x 256-col output tile.
// Each wave owns 32 cols and all 64 rows: 4 row-tiles x 2 col-tiles = 8
// accumulators, K-loop in steps of 4 using V_WMMA_F32_16X16X4_F32 (fp32 path).
// Each B fragment feeds 4 WMMAs; each A fragment feeds 2.
// Bounds handling is hoisted to one uniform branch (only the last block of
// 1563 is partial), so the hot epilogue is pure immediate-offset stores.
//
// f32 16x4 A fragment  (per ISA): lanes 0-15: {K=k0, k0+1} of row M=lane;
//                                 lanes 16-31: {K=k0+2, k0+3} of row M=lane-16.
// f32 4x16 B fragment  (mirrored): lanes 0-15: col N=lane, rows {k0, k0+1};
//                                  lanes 16-31: col N=lane-16, rows {k0+2, k0+3}.
// f32 16x16 C/D: VGPR r: lanes 0-15 -> M=r, N=lane; lanes 16-31 -> M=r+8.
// ---------------------------------------------------------------------------
template <int K, bool RELU>
__global__ __launch_bounds__(256) void gemm_bias_act_kernel(
    const float* __restrict__ A, const float* __restrict__ W,
    const float* __restrict__ bias, float* __restrict__ out, int nRows) {
    constexpr int ROWS = 64;                   // rows per block
    constexpr int LDA = K + 4;                 // pad: bank-conflict-free column reads
    __shared__ float As[ROWS * LDA];

    const int tid = threadIdx.x;
    const int rowBase = blockIdx.x * ROWS;
    const bool fullTile = (rowBase + ROWS) <= nRows;   // uniform across block

    // ---- stage 64 x K A-tile into LDS (float4, coalesced) ----
    constexpr int NF4 = ROWS * K / 4;
    constexpr int CPR = K / 4;                 // float4 per row
    if (fullTile) {
        const float4* ap = (const float4*)(A + (size_t)rowBase * K);
#pragma unroll
        for (int i = tid; i < NF4; i += 256) {
            int r = i / CPR;
            int cq = i - r * CPR;
            *(float4*)(&As[r * LDA + cq * 4]) = ap[i];
        }
    } else {
#pragma unroll
        for (int i = tid; i < NF4; i += 256) {
            int r = i / CPR;
            int cq = i - r * CPR;
            int row = rowBase + r;
            float4 v = make_float4(0.f, 0.f, 0.f, 0.f);
            if (row < nRows) v = *(const float4*)(A + (size_t)row * K + cq * 4);
            *(float4*)(&As[r * LDA + cq * 4]) = v;
        }
    }
    __syncthreads();

    const int lane = tid & 31;
    const int wave = tid >> 5;
    const int half = lane >> 4;                // 0: K pair {k0,k0+1}; 1: {k0+2,k0+3}
    const int l16  = lane & 15;
    const int colBase = wave * 32;
    const int col0 = colBase + l16;
    const int col1 = col0 + 16;

    v8f acc[4][2];
#pragma unroll
    for (int mt = 0; mt < 4; mt++)
#pragma unroll
        for (int nt = 0; nt < 2; nt++) acc[mt][nt] = (v8f){0.f, 0.f, 0.f, 0.f, 0.f, 0.f, 0.f, 0.f};

#pragma unroll 2
    for (int k0 = 0; k0 < K; k0 += 4) {
        const int ka = k0 + half * 2;
        v2f a[4];
#pragma unroll
        for (int mt = 0; mt < 4; mt++) {
            const float* ap = &As[(l16 + 16 * mt) * LDA + ka];
            a[mt].x = ap[0];
            a[mt].y = ap[1];
        }
        const float* wp = W + (size_t)ka * DHID;
        v2f b0, b1;
        b0.x = wp[col0];
        b0.y = wp[DHID + col0];
        b1.x = wp[col1];
        b1.y = wp[DHID + col1];
#pragma unroll
        for (int mt = 0; mt < 4; mt++) {
            acc[mt][0] = __builtin_amdgcn_wmma_f32_16x16x4_f32(false, a[mt], false, b0, (short)0,
                                                               acc[mt][0], false, false);
            acc[mt][1] = __builtin_amdgcn_wmma_f32_16x16x4_f32(false, a[mt], false, b1, (short)0,
                                                               acc[mt][1], false, false);
        }
    }

    const float bc0 = bias[col0];
    const float bc1 = bias[col1];
    if (fullTile) {
        // fast path: one base pointer, all offsets fold into the 24-bit IOFFSET
        float* op = out + (size_t)(rowBase + half * 8) * DHID;
#pragma unroll
        for (int mt = 0; mt < 4; mt++) {
#pragma unroll
            for (int r = 0; r < 8; r++) {
                float v0 = acc[mt][0][r] + bc0;
                float v1 = acc[mt][1][r] + bc1;
                if (RELU) {
                    v0 = fmaxf(v0, 0.f);
                    v1 = fmaxf(v1, 0.f);
                }
                op[(mt * 16 + r) * DHID + col0] = v0;
                op[(mt * 16 + r) * DHID + col1] = v1;
            }
        }
    } else {
#pragma unroll
        for (int mt = 0; mt < 4; mt++) {
#pragma unroll
            for (int r = 0; r < 8; r++) {
                int row = rowBase + mt * 16 + r + half * 8;
                if (row >= nRows) continue;
                float v0 = acc[mt][0][r] + bc0;
                float v1 = acc[mt][1][r] + bc1;
                if (RELU) {
                    v0 = fmaxf(v0, 0.f);
                    v1 = fmaxf(v1, 0.f);
                }
                out[(size_t)row * DHID + col0] = v0;
                out[(size_t)row * DHID + col1] = v1;
            }
        }
    }
}

// ---------------------------------------------------------------------------
extern "C" void kernel_launch(void* const* d_in, const int* in_sizes, int n_in,
                              void* d_out, int out_size, void* d_ws, size_t ws_size,
                              hipStream_t stream) {
    const float* x   = (const float*)d_in[0];
    const int*   ei  = (const int*)d_in[1];
    const float* w11 = (const float*)d_in[2];
    const float* b11 = (const float*)d_in[3];
    const float* w12 = (const float*)d_in[4];
    const float* b12 = (const float*)d_in[5];
    const float* w21 = (const float*)d_in[6];
    const float* b21 = (const float*)d_in[7];
    const float* w22 = (const float*)d_in[8];
    const float* b22 = (const float*)d_in[9];
    const float* w31 = (const float*)d_in[10];
    const float* b31 = (const float*)d_in[11];
    const float* w32 = (const float*)d_in[12];
    const float* b32 = (const float*)d_in[13];

    const int N = in_sizes[0] / 128;           // DIN = 128
    const int E = in_sizes[1] / 2;
    const int* src = ei;                       // edge_index row 0
    const int* dst = ei + E;                   // edge_index row 1
    float* out = (float*)d_out;

    float* agg = (float*)d_ws;                 // N x 256 (layer1 uses N x 128 prefix)
    float* T   = agg + (size_t)N * DHID;       // N x 256
    float* H   = T + (size_t)N * DHID;         // N x 256

    const dim3 blk(256);
    auto cdiv = [](long a, long b) { return (unsigned)((a + b - 1) / b); };
    const unsigned gRows = cdiv(N, 64);

    // ---------------- layer 1 (D = 128 -> 256) ----------------
    copy_f4_kernel<<<cdiv((long)N * 128 / 4, 256), blk, 0, stream>>>(
        (const float4*)x, (float4*)agg, (long)N * 128 / 4);
    scatter_add_kernel<128><<<cdiv((long)E * 32, 256), blk, 0, stream>>>(x, src, dst, agg, E);
    gemm_bias_act_kernel<128, true><<<gRows, blk, 0, stream>>>(agg, w11, b11, T, N);
    gemm_bias_act_kernel<256, true><<<gRows, blk, 0, stream>>>(T, w12, b12, H, N);

    // ---------------- layer 2 (256 -> 256) ----------------
    copy_f4_kernel<<<cdiv((long)N * 256 / 4, 256), blk, 0, stream>>>(
        (const float4*)H, (float4*)agg, (long)N * 256 / 4);
    scatter_add_kernel<256><<<cdiv((long)E * 64, 256), blk, 0, stream>>>(H, src, dst, agg, E);
    gemm_bias_act_kernel<256, true><<<gRows, blk, 0, stream>>>(agg, w21, b21, T, N);
    gemm_bias_act_kernel<256, true><<<gRows, blk, 0, stream>>>(T, w22, b22, H, N);

    // ---------------- layer 3 (256 -> 256, no trailing relu) ----------------
    copy_f4_kernel<<<cdiv((long)N * 256 / 4, 256), blk, 0, stream>>>(
        (const float4*)H, (float4*)agg, (long)N * 256 / 4);
    scatter_add_kernel<256><<<cdiv((long)E * 64, 256), blk, 0, stream>>>(H, src, dst, agg, E);
    gemm_bias_act_kernel<256, true><<<gRows, blk, 0, stream>>>(agg, w31, b31, T, N);
    gemm_bias_act_kernel<256, false><<<gRows, blk, 0, stream>>>(T, w32, b32, out, N);
}